// DistributedMPNN_18846316495505
// MI455X (gfx1250) — compile-verified
//
#include <hip/hip_runtime.h>
#include <hip/hip_bf16.h>

typedef float v2f __attribute__((ext_vector_type(2)));
typedef float v8f __attribute__((ext_vector_type(8)));

// D = A(16x4 f32) * B(4x16 f32) + C(16x16 f32)
#define WMMA_F32(a, b, c) \
  __builtin_amdgcn_wmma_f32_16x16x4_f32(false, (a), false, (b), (short)0, (c), false, false)

constexpr int kWavesPerBlock = 8;    // 256 threads = 8 wave32
constexpr int kIn1Stride = 13;       // 10 feats + 2 zero-pad + 1 bank pad
constexpr int kH1Stride  = 33;       // 32 + 1 bank pad
constexpr int kWaveLds   = 16 * kIn1Stride + 16 * kH1Stride + 32; // + 32 src/dst ids

// ---------------------------------------------------------------------------
// Edge MLP: m = relu(relu([x[src]||ea] @ Wm1 + bm1) @ Wm2 + bm2); atomic scatter
// Each wave loops over 16-edge tiles with B operands resident in VGPRs.
// ---------------------------------------------------------------------------
__global__ __launch_bounds__(256) void mpnn_edge_kernel(
    const float* __restrict__ h,    // [N,9]
    const int*   __restrict__ src,  // [E]
    const int*   __restrict__ dst,  // [E]
    const float* __restrict__ ea,   // [E]
    const float* __restrict__ Wm1, const float* __restrict__ bm1,   // [10,32],[32]
    const float* __restrict__ Wm2, const float* __restrict__ bm2,   // [32,32],[32]
    float* __restrict__ aggr,       // [N,32]
    int nEdges, int nTiles, int tileStride, int iters)
{
  __shared__ float lds[kWavesPerBlock * kWaveLds];
  const int wave = threadIdx.x >> 5;
  const int lane = threadIdx.x & 31;
  const int m = lane & 15;    // row/col within 16-tile
  const int g = lane >> 4;    // lane half (selects K/M offset per ISA layout)

  float* in1  = lds + wave * kWaveLds;       // [16][13] staged layer-1 input
  float* h1s  = in1 + 16 * kIn1Stride;       // [16][33] layer-1 activations
  int*   sIdx = (int*)(h1s + 16 * kH1Stride); // [32]: 0..15 = src, 16..31 = dst

  // ---- Hoist loop-invariant B operands + biases into registers ----
  // B layout (4x16 f32): lane half g, component c -> K = c + 2*g (+4*ks), N = m (+16*tile)
  v2f B1[3][2];
#pragma unroll
  for (int ks = 0; ks < 3; ++ks) {
    const int kk = ks * 4 + 2 * g;
    B1[ks][0].x = (kk     < 10) ? Wm1[(kk    ) * 32 + m]      : 0.0f;
    B1[ks][0].y = (kk + 1 < 10) ? Wm1[(kk + 1) * 32 + m]      : 0.0f;
    B1[ks][1].x = (kk     < 10) ? Wm1[(kk    ) * 32 + 16 + m] : 0.0f;
    B1[ks][1].y = (kk + 1 < 10) ? Wm1[(kk + 1) * 32 + 16 + m] : 0.0f;
  }
  v2f B2[8][2];
#pragma unroll
  for (int ks = 0; ks < 8; ++ks) {
    const int kk = ks * 4 + 2 * g;
    B2[ks][0].x = Wm2[(kk    ) * 32 + m];
    B2[ks][0].y = Wm2[(kk + 1) * 32 + m];
    B2[ks][1].x = Wm2[(kk    ) * 32 + 16 + m];
    B2[ks][1].y = Wm2[(kk + 1) * 32 + 16 + m];
  }
  const float cb0 = bm1[m], cb1 = bm1[16 + m];
  const float db0 = bm2[m], db1 = bm2[16 + m];

  // Zero the K=10,11 pad columns once; the gather loop never touches them.
  in1[m * kIn1Stride + 10 + g] = 0.0f;

  const int waveId = blockIdx.x * kWavesPerBlock + wave;

  for (int it = 0; it < iters; ++it) {
    const int tile = waveId + it * tileStride;
    const bool active = tile < nTiles;
    const int base = (active ? tile : 0) * 16;

    // Stage src/dst ids for the 16 edges (lane half 0 -> src, half 1 -> dst).
    {
      int eid = base + m;
      if (eid >= nEdges) eid = nEdges - 1;
      sIdx[lane] = (g == 0) ? src[eid] : dst[eid];
    }
    __threadfence_block();   // wave-private LDS: order DS ops (HW keeps per-wave DS in order)

    // Stage [16 edges][10 feats] = x[src](9) ++ edge_attr(1).
#pragma unroll
    for (int i = lane; i < 160; i += 32) {
      const int e = i / 10, f = i % 10;
      float v;
      if (f < 9) {
        v = h[(size_t)sIdx[e] * 9 + f];
      } else {
        int eid = base + e;
        if (eid >= nEdges) eid = nEdges - 1;
        v = ea[eid];
      }
      in1[e * kIn1Stride + f] = v;
    }
    __threadfence_block();

    // ---- Layer 1: [16,12] @ [12,32] + bm1 ----
    v8f c0, c1;
#pragma unroll
    for (int r = 0; r < 8; ++r) { c0[r] = cb0; c1[r] = cb1; }
#pragma unroll
    for (int ks = 0; ks < 3; ++ks) {
      const int kk = ks * 4 + 2 * g;
      v2f a;
      a.x = in1[m * kIn1Stride + kk];
      a.y = in1[m * kIn1Stride + kk + 1];
      c0 = WMMA_F32(a, B1[ks][0], c0);
      c1 = WMMA_F32(a, B1[ks][1], c1);
    }
    // relu + spill to LDS so layer-2 re-reads in A-operand layout
#pragma unroll
    for (int r = 0; r < 8; ++r) {
      const int row = r + 8 * g;  // C layout: M = vgpr + 8*lane_group
      h1s[row * kH1Stride + m]      = fmaxf(c0[r], 0.0f);
      h1s[row * kH1Stride + 16 + m] = fmaxf(c1[r], 0.0f);
    }
    __threadfence_block();

    // ---- Layer 2: [16,32] @ [32,32] + bm2 ----
    v8f d0, d1;
#pragma unroll
    for (int r = 0; r < 8; ++r) { d0[r] = db0; d1[r] = db1; }
#pragma unroll
    for (int ks = 0; ks < 8; ++ks) {
      const int kk = ks * 4 + 2 * g;
      v2f a;
      a.x = h1s[m * kH1Stride + kk];
      a.y = h1s[m * kH1Stride + kk + 1];
      d0 = WMMA_F32(a, B2[ks][0], d0);
      d1 = WMMA_F32(a, B2[ks][1], d1);
    }

    // relu + scatter-sum into aggr[dst] (aggr = 6.4 MB -> L2-resident atomics)
#pragma unroll
    for (int r = 0; r < 8; ++r) {
      const int row = r + 8 * g;
      if (active && (base + row) < nEdges) {
        const size_t nb = (size_t)sIdx[16 + row] * 32;
        atomicAdd(&aggr[nb + m],      fmaxf(d0[r], 0.0f));
        atomicAdd(&aggr[nb + 16 + m], fmaxf(d1[r], 0.0f));
      }
    }
    __threadfence_block();   // keep staging of next tile behind this tile's LDS reads
  }
}

// ---------------------------------------------------------------------------
// Node update: h' = [x0 || relu(relu([x||aggr] @ Wu1 + bu1) @ Wu2 + bu2)]
// ---------------------------------------------------------------------------
__global__ __launch_bounds__(256) void mpnn_node_kernel(
    const float* __restrict__ h, const float* __restrict__ aggr,
    const float* __restrict__ Wu1, const float* __restrict__ bu1,  // [41,16],[16]
    const float* __restrict__ Wu2, const float* __restrict__ bu2,  // [16,8],[8]
    float* __restrict__ hout, int nNodes)
{
  __shared__ float sW1[41 * 16];
  __shared__ float sb1[16];
  __shared__ float sW2[16 * 8];
  __shared__ float sb2[8];
  for (int i = threadIdx.x; i < 41 * 16; i += blockDim.x) sW1[i] = Wu1[i];
  for (int i = threadIdx.x; i < 16 * 8;  i += blockDim.x) sW2[i] = Wu2[i];
  if (threadIdx.x < 16) sb1[threadIdx.x] = bu1[threadIdx.x];
  if (threadIdx.x < 8)  sb2[threadIdx.x] = bu2[threadIdx.x];
  __syncthreads();

  const int n = blockIdx.x * blockDim.x + threadIdx.x;
  if (n >= nNodes) return;

  float in[41];
#pragma unroll
  for (int j = 0; j < 9; ++j)  in[j]     = h[(size_t)n * 9 + j];
#pragma unroll
  for (int j = 0; j < 32; ++j) in[9 + j] = aggr[(size_t)n * 32 + j];

  float t[16];
#pragma unroll
  for (int o = 0; o < 16; ++o) {
    float s = sb1[o];
#pragma unroll
    for (int k = 0; k < 41; ++k) s = fmaf(in[k], sW1[k * 16 + o], s);
    t[o] = fmaxf(s, 0.0f);
  }
  hout[(size_t)n * 9] = in[0];   // keep state channel
#pragma unroll
  for (int o = 0; o < 8; ++o) {
    float s = sb2[o];
#pragma unroll
    for (int k = 0; k < 16; ++k) s = fmaf(t[k], sW2[k * 8 + o], s);
    hout[(size_t)n * 9 + 1 + o] = fmaxf(s, 0.0f);
  }
}

// ---------------------------------------------------------------------------
// Head: sigmoid(relu(h[:,1:] @ Wh1 + bh1) @ Wh2 + bh2)
// ---------------------------------------------------------------------------
__global__ __launch_bounds__(256) void mpnn_h2o_kernel(
    const float* __restrict__ h,
    const float* __restrict__ Wh1, const float* __restrict__ bh1,  // [8,16],[16]
    const float* __restrict__ Wh2, const float* __restrict__ bh2,  // [16,1],[1]
    float* __restrict__ out, int nNodes)
{
  const int n = blockIdx.x * blockDim.x + threadIdx.x;
  if (n >= nNodes) return;
  float e[8];
#pragma unroll
  for (int j = 0; j < 8; ++j) e[j] = h[(size_t)n * 9 + 1 + j];
  float acc = bh2[0];
#pragma unroll
  for (int o = 0; o < 16; ++o) {
    float s = bh1[o];
#pragma unroll
    for (int k = 0; k < 8; ++k) s = fmaf(e[k], Wh1[k * 16 + o], s);
    acc = fmaf(fmaxf(s, 0.0f), Wh2[o], acc);
  }
  out[n] = 1.0f / (1.0f + expf(-acc));
}

__global__ void mpnn_zero_kernel(float* __restrict__ p, size_t n) {
  size_t i = (size_t)blockIdx.x * blockDim.x + threadIdx.x;
  if (i < n) p[i] = 0.0f;
}

// ---------------------------------------------------------------------------
extern "C" void kernel_launch(void* const* d_in, const int* in_sizes, int n_in,
                              void* d_out, int out_size, void* d_ws, size_t ws_size,
                              hipStream_t stream) {
  const float* x   = (const float*)d_in[0];
  const int*   ei  = (const int*)d_in[1];
  const float* ea  = (const float*)d_in[2];
  const float* Wm1 = (const float*)d_in[3];
  const float* bm1 = (const float*)d_in[4];
  const float* Wm2 = (const float*)d_in[5];
  const float* bm2 = (const float*)d_in[6];
  const float* Wu1 = (const float*)d_in[7];
  const float* bu1 = (const float*)d_in[8];
  const float* Wu2 = (const float*)d_in[9];
  const float* bu2 = (const float*)d_in[10];
  const float* Wh1 = (const float*)d_in[11];
  const float* bh1 = (const float*)d_in[12];
  const float* Wh2 = (const float*)d_in[13];
  const float* bh2 = (const float*)d_in[14];

  const int N = in_sizes[0] / 9;
  const int E = in_sizes[1] / 2;
  const int* src = ei;
  const int* dst = ei + E;

  // workspace: two h ping-pong buffers [N,9] + aggr [N,32]
  float* hb0  = (float*)d_ws;
  float* hb1  = hb0 + (size_t)N * 9;
  float* aggr = hb1 + (size_t)N * 9;

  const int nTiles = (E + 15) / 16;
  // ~8 tiles per wave: blocks = ceil(nTiles / 64), 8 waves per block
  int edgeBlocks = (nTiles + 63) / 64;
  if (edgeBlocks < 1) edgeBlocks = 1;
  const int tileStride = edgeBlocks * kWavesPerBlock;        // total waves
  const int iters = (nTiles + tileStride - 1) / tileStride;  // uniform per-wave trip count

  const int nodeBlocks = (N + 255) / 256;
  const size_t aggrN   = (size_t)N * 32;
  const int zeroBlocks = (int)((aggrN + 255) / 256);

  const float* hcur = x;
  float* bufs[2] = {hb0, hb1};
  for (int r = 0; r < 3; ++r) {
    mpnn_zero_kernel<<<zeroBlocks, 256, 0, stream>>>(aggr, aggrN);
    mpnn_edge_kernel<<<edgeBlocks, 256, 0, stream>>>(hcur, src, dst, ea,
                                                     Wm1, bm1, Wm2, bm2, aggr,
                                                     E, nTiles, tileStride, iters);
    mpnn_node_kernel<<<nodeBlocks, 256, 0, stream>>>(hcur, aggr, Wu1, bu1, Wu2, bu2,
                                                     bufs[r & 1], N);
    hcur = bufs[r & 1];
  }
  mpnn_h2o_kernel<<<nodeBlocks, 256, 0, stream>>>(hcur, Wh1, bh1, Wh2, bh2,
                                                  (float*)d_out, N);
}